// Network_54073638257187
// MI455X (gfx1250) — compile-verified
//
#include <hip/hip_runtime.h>

// ---------------------------------------------------------------------------
// pred[b,j] = sigmoid( sum_k softmax_k(u)_k * v_k + emo_b ),
//   u = H_row . fc_w  (fc_b cancels in softmax),  v = H_row . emo_w
// Stage 1: dual GEMV over all B*T=131072 rows via V_WMMA_F32_16X16X4_F32.
//   K-order permuted so each lane streams a contiguous half-row:
//   step s: K0,K1 = features 2s,2s+1 (lanes 0-15), K2,K3 = 384+2s,384+2s+1
//   (lanes 16-31). One b128 load feeds two WMMAs.
// Stage 2: per-clause masked softmax + weighted sum + sigmoid (wave32).
// HBM-bound: one pass over 402 MB -> ~17 us at 23.3 TB/s.
// ---------------------------------------------------------------------------

typedef __attribute__((ext_vector_type(2))) float v2f;
typedef __attribute__((ext_vector_type(4))) float v4f;
typedef __attribute__((ext_vector_type(8))) float v8f;

#define B_   32
#define T_   4096
#define D_   768
#define DH_  384   // half feature dim
#define JMAX 64

__global__ __launch_bounds__(256) void gemv2_wmma_kernel(
    const float* __restrict__ H, const float* __restrict__ fcw,
    const float* __restrict__ emow, float2* __restrict__ uv) {
  __shared__ __align__(16) float s_fcw[D_];
  __shared__ __align__(16) float s_emow[D_];
  __shared__ __align__(16) float s_zero[4];
  for (int i = threadIdx.x; i < D_; i += 256) {
    s_fcw[i]  = fcw[i];
    s_emow[i] = emow[i];
  }
  if (threadIdx.x < 4) s_zero[threadIdx.x] = 0.0f;
  __syncthreads();

  const int lane = threadIdx.x & 31;
  const int wave = threadIdx.x >> 5;
  const int tile = blockIdx.x * 8 + wave;    // 8192 tiles, 16 rows each
  const int m    = lane & 15;                // A row / B column (token) index
  const int h    = lane >> 4;                // 0: features 0..383, 1: 384..767

  // B stream: token row m, contiguous half-row [h*384, h*384+384)
  const size_t row = (size_t)tile * 16 + m;
  const float* bp  = H + row * D_ + h * DH_;     // 16B aligned

  // A stream: fc_w for m==0, emo_w for m==1, zeros otherwise (stride 0)
  const float* ap = (m == 0) ? (s_fcw + h * DH_)
                  : (m == 1) ? (s_emow + h * DH_)
                             : s_zero;
  const int astep = (m < 2) ? 4 : 0;

  v8f c = {};
#pragma unroll 4
  for (int t = 0; t < DH_ / 4; ++t) {            // 96 iters, 2 WMMA steps each
    v4f b4 = *(const v4f*)(bp + 4 * t);
    v4f a4 = *(const v4f*)ap;
    ap += astep;
    v2f a0 = __builtin_shufflevector(a4, a4, 0, 1);
    v2f b0 = __builtin_shufflevector(b4, b4, 0, 1);
    c = __builtin_amdgcn_wmma_f32_16x16x4_f32(false, a0, false, b0,
                                              (short)0, c, false, false);
    v2f a1 = __builtin_shufflevector(a4, a4, 2, 3);
    v2f b1 = __builtin_shufflevector(b4, b4, 2, 3);
    c = __builtin_amdgcn_wmma_f32_16x16x4_f32(false, a1, false, b1,
                                              (short)0, c, false, false);
  }

  if (lane < 16) {  // D row0 = u, row1 = v, N = lane
    uv[(size_t)tile * 16 + lane] = make_float2(c[0], c[1]);
  }
}

// ---- Stage 2: one wave per clause (b,j): masked softmax(u), weighted v ----
__global__ __launch_bounds__(256) void softmax_head_kernel(
    const float2* __restrict__ uv, const int* __restrict__ clause_len,
    const float* __restrict__ emo_b, float* __restrict__ out) {
  const int lane = threadIdx.x & 31;
  const int cj   = blockIdx.x * 8 + (threadIdx.x >> 5);  // 2048 clauses
  const int b    = cj >> 6;
  const int j    = cj & 63;

  const int* cl = clause_len + b * JMAX;
  int start = 0;
  for (int t = 0; t < j; ++t) start += cl[t];   // exclusive prefix sum
  const int len = cl[j];                        // in [1, 64]

  const float2* p = uv + (size_t)b * T_ + start;
  float u1 = -3.0e38f, v1 = 0.0f, u2 = -3.0e38f, v2 = 0.0f;
  if (lane < len)      { float2 a = p[lane];      u1 = a.x; v1 = a.y; }
  if (lane + 32 < len) { float2 a = p[lane + 32]; u2 = a.x; v2 = a.y; }

  float mx = fmaxf(u1, u2);
#pragma unroll
  for (int off = 16; off > 0; off >>= 1)
    mx = fmaxf(mx, __shfl_xor(mx, off, 32));

  float e1 = (lane < len)      ? expf(u1 - mx) : 0.0f;
  float e2 = (lane + 32 < len) ? expf(u2 - mx) : 0.0f;
  float se = e1 + e2;
  float wv = e1 * v1 + e2 * v2;
#pragma unroll
  for (int off = 16; off > 0; off >>= 1) {
    se += __shfl_xor(se, off, 32);
    wv += __shfl_xor(wv, off, 32);
  }

  if (lane == 0) {
    float x = wv / se + emo_b[0];
    out[cj] = 1.0f / (1.0f + expf(-x));
  }
}

extern "C" void kernel_launch(void* const* d_in, const int* in_sizes, int n_in,
                              void* d_out, int out_size, void* d_ws, size_t ws_size,
                              hipStream_t stream) {
  const float* H          = (const float*)d_in[0];  // [32,4096,768] f32
  const int*   clause_len = (const int*)d_in[1];    // [32,64] i32
  const float* fc_w       = (const float*)d_in[2];  // [768]
  // d_in[3] = fc_b: uniform logit shift, cancels in softmax -> unused
  const float* emo_w      = (const float*)d_in[4];  // [768]
  const float* emo_b      = (const float*)d_in[5];  // [1]
  float*  out = (float*)d_out;                      // [32,64]
  float2* uv  = (float2*)d_ws;                      // B*T float2 = 1 MB scratch

  gemv2_wmma_kernel<<<1024, 256, 0, stream>>>(H, fc_w, emo_w, uv);
  softmax_head_kernel<<<256, 256, 0, stream>>>(uv, clause_len, emo_b, out);
}